// ContextualLoss_14353780703730
// MI455X (gfx1250) — compile-verified
//
#include <hip/hip_runtime.h>

typedef float v2f __attribute__((ext_vector_type(2)));
typedef float v8f __attribute__((ext_vector_type(8)));

#define NSP 6400      // H*W
#define NT  400       // 6400 / 16 tiles
#define NB  2         // batch
#define HINV 10.0f    // 1 / H_BANDWIDTH
#define NEGBIG -3.0e38f
#define OPPAD 4       // extra tile records so prefetch never reads out of bounds

// ---------------------------------------------------------------- means ----
__global__ void cl_mean_kernel(const float* __restrict__ target,
                               float* __restrict__ meanp) {
    __shared__ float sh[256];
    const int bc = blockIdx.x;                 // 0..3  (b*2 + c)
    const float* t = target + (size_t)bc * NSP;
    float s = 0.0f;
    for (int i = threadIdx.x; i < NSP; i += 256) s += t[i];
    sh[threadIdx.x] = s;
    __syncthreads();
    for (int off = 128; off > 0; off >>= 1) {
        if ((int)threadIdx.x < off) sh[threadIdx.x] += sh[threadIdx.x + off];
        __syncthreads();
    }
    if (threadIdx.x == 0) meanp[bc] = sh[0] * (1.0f / 6400.0f);
}

// ------------------------------------------------------------ normalize ----
// Writes tile-ready WMMA operands: per 16-tile, float2[32] records where
// lanes 0-15 carry {ch0, ch1} (K=0,1) and lanes 16-31 carry zeros (K=2,3).
__global__ void cl_normalize_kernel(const float* __restrict__ pred,
                                    const float* __restrict__ target,
                                    const float* __restrict__ meanp,
                                    float2* __restrict__ xop,
                                    float2* __restrict__ yop) {
    int idx = blockIdx.x * blockDim.x + threadIdx.x;   // b*NSP + n
    if (idx >= NB * NSP) return;
    int b = idx / NSP, n = idx - b * NSP;
    float m0 = meanp[b * 2 + 0], m1 = meanp[b * 2 + 1];

    float x0 = pred[(size_t)(b * 2 + 0) * NSP + n] - m0;
    float x1 = pred[(size_t)(b * 2 + 1) * NSP + n] - m1;
    float ix = 1.0f / (sqrtf(x0 * x0 + x1 * x1) + 1e-10f);
    float y0 = target[(size_t)(b * 2 + 0) * NSP + n] - m0;
    float y1 = target[(size_t)(b * 2 + 1) * NSP + n] - m1;
    float iy = 1.0f / (sqrtf(y0 * y0 + y1 * y1) + 1e-10f);

    size_t p = ((size_t)b * NT + (n >> 4)) * 32 + (n & 15);
    float2 z; z.x = 0.0f; z.y = 0.0f;
    float2 xv; xv.x = x0 * ix; xv.y = x1 * ix;
    float2 yv; yv.x = y0 * iy; yv.y = y1 * iy;
    xop[p] = xv;  xop[p + 16] = z;
    yop[p] = yv;  yop[p + 16] = z;
}

__device__ __forceinline__ v2f as_v2f(float2 f) {
    v2f v; v.x = f.x; v.y = f.y; return v;
}

// -------------------------------------------------- pass 1: row max(dot) ----
// Block = 16 rows n (one x-tile); 4 waves split the m reduction stride-4.
// Tile: D[M=m, N=n] = y[m].x[n]; reduce over M (VGPR axis) + lane^16 + LDS.
__global__ void cl_rowstats_kernel(const float2* __restrict__ xop,
                                   const float2* __restrict__ yop,
                                   float* __restrict__ cq) {
    __shared__ float red[64];
    const int tile  = blockIdx.x;            // 0..799
    const int b     = tile / NT;
    const int tt    = tile - b * NT;
    const int tbase = tt * 16;
    const int warp  = threadIdx.x >> 5;
    const int lane  = threadIdx.x & 31;
    const size_t ob = (size_t)b * NT;

    v2f bv = as_v2f(xop[(ob + tt) * 32 + lane]);

    v8f vmax;
#pragma unroll
    for (int i = 0; i < 8; ++i) vmax[i] = NEGBIG;

    const float2* ap = yop + (ob + warp) * 32 + lane;
    float2 cur = ap[0];
#pragma unroll 2
    for (int it = 0; it < NT / 4; ++it) {
        float2 nxt = ap[128];                 // next tile for this warp (in-bounds: padded)
        v8f cz = {};
        v8f d = __builtin_amdgcn_wmma_f32_16x16x4_f32(false, as_v2f(cur), false, bv,
                                                      (short)0, cz, false, false);
#pragma unroll
        for (int i = 0; i < 8; ++i) vmax[i] = fmaxf(vmax[i], d[i]);
        cur = nxt;
        ap += 128;
    }
    float mx = vmax[0];
#pragma unroll
    for (int i = 1; i < 8; ++i) mx = fmaxf(mx, vmax[i]);
    mx = fmaxf(mx, __shfl_xor(mx, 16, 32));
    if (lane < 16) red[warp * 16 + lane] = mx;
    __syncthreads();
    if (threadIdx.x < 16) {
        int l = threadIdx.x;
        float mm = fmaxf(fmaxf(red[l], red[16 + l]), fmaxf(red[32 + l], red[48 + l]));
        int n = tbase + l;
        float R    = (1.0f - mm) + 1e-5f;    // rmin + EPS_MIN
        float invR = 1.0f / R;
        cq[2 * (b * NSP + n) + 0] = invR * HINV;           // c = 1/(h*R)
        cq[2 * (b * NSP + n) + 1] = (1.0f - invR) * HINV;  // k = (1 - 1/R)/h
    }
}

// ------------------------------------------- pass 2: row sum of weights ----
__global__ void cl_rowsum_kernel(const float2* __restrict__ xop,
                                 const float2* __restrict__ yop,
                                 float* __restrict__ cq) {
    __shared__ float red[64];
    const int tile  = blockIdx.x;
    const int b     = tile / NT;
    const int tt    = tile - b * NT;
    const int tbase = tt * 16;
    const int warp  = threadIdx.x >> 5;
    const int lane  = threadIdx.x & 31;
    const size_t ob = (size_t)b * NT;
    const int  col  = tbase + (lane & 15);

    v2f bv = as_v2f(xop[(ob + tt) * 32 + lane]);
    const float cn = cq[2 * (b * NSP + col) + 0];
    const float kn = cq[2 * (b * NSP + col) + 1];

    v8f acc = {};
    const float2* ap = yop + (ob + warp) * 32 + lane;
    float2 cur = ap[0];
#pragma unroll 2
    for (int it = 0; it < NT / 4; ++it) {
        float2 nxt = ap[128];
        v8f cz = {};
        v8f d = __builtin_amdgcn_wmma_f32_16x16x4_f32(false, as_v2f(cur), false, bv,
                                                      (short)0, cz, false, false);
#pragma unroll
        for (int i = 0; i < 8; ++i) acc[i] += __expf(fmaf(cn, d[i], kn));
        cur = nxt;
        ap += 128;
    }
    float s = acc[0];
#pragma unroll
    for (int i = 1; i < 8; ++i) s += acc[i];
    s += __shfl_xor(s, 16, 32);
    if (lane < 16) red[warp * 16 + lane] = s;
    __syncthreads();
    if (threadIdx.x < 16) {
        int l = threadIdx.x;
        float rs = (red[l] + red[16 + l]) + (red[32 + l] + red[48 + l]);
        // threadIdx < 16 => warp0, lane<16: kn in-register matches col tbase+l
        cq[2 * (b * NSP + tbase + l) + 1] = kn - __logf(rs);   // q = k - log(rowsum)
    }
}

// ---------------------------- pass 3: column max over n (log domain) ----
// Block = 16 columns m (one y-tile); 4 waves split the n reduction stride-4.
// Tile: D[M=n, N=m] = x[n].y[m]; reduce over M (VGPR axis) + lane^16 + LDS.
__global__ void cl_colmax_kernel(const float2* __restrict__ xop,
                                 const float2* __restrict__ yop,
                                 const float* __restrict__ cq,
                                 float* __restrict__ colpart) {
    __shared__ float red[64];
    const int tile  = blockIdx.x;            // b*NT + mtile
    const int b     = tile / NT;
    const int tt    = tile - b * NT;
    const int warp  = threadIdx.x >> 5;
    const int lane  = threadIdx.x & 31;
    const size_t ob = (size_t)b * NT;

    v2f bv = as_v2f(yop[(ob + tt) * 32 + lane]);
    // (c,q) pairs viewed as float4: two n-slots per load
    const float4* ckp = (const float4*)cq + (size_t)b * (NSP / 2) + warp * 8
                      + ((lane < 16) ? 0 : 4);

    float mx = NEGBIG;
    const float2* ap = xop + (ob + warp) * 32 + lane;
    float2 cur = ap[0];
#pragma unroll 2
    for (int it = 0; it < NT / 4; ++it) {
        float2 nxt = ap[128];
        v8f cz = {};
        v8f d = __builtin_amdgcn_wmma_f32_16x16x4_f32(false, as_v2f(cur), false, bv,
                                                      (short)0, cz, false, false);
#pragma unroll
        for (int j = 0; j < 4; ++j) {
            float4 ck = ckp[j];                        // (c, q, c, q)
            mx = fmaxf(mx, fmaf(ck.x, d[2 * j],     ck.y));
            mx = fmaxf(mx, fmaf(ck.z, d[2 * j + 1], ck.w));
        }
        cur = nxt;
        ap  += 128;
        ckp += 32;                                     // 4 tiles * 8 float4
    }
    mx = fmaxf(mx, __shfl_xor(mx, 16, 32));   // full max per m within this wave's chunk
    if (lane < 16) red[warp * 16 + lane] = mx;
    __syncthreads();
    if (warp == 0) {
        float mm = NEGBIG;
        if (lane < 16)
            mm = fmaxf(fmaxf(red[lane], red[16 + lane]),
                       fmaxf(red[32 + lane], red[48 + lane]));
        float e = (lane < 16) ? __expf(mm) : 0.0f;
#pragma unroll
        for (int off = 16; off > 0; off >>= 1) e += __shfl_xor(e, off, 32);
        if (lane == 0) colpart[tile] = e;     // sum of 16 column maxima
    }
}

// -------------------------------------------------------------- finalize ----
__global__ void cl_finalize_kernel(const float* __restrict__ colpart,
                                   float* __restrict__ out) {
    __shared__ float sh0[128], sh1[128];
    int t = threadIdx.x;
    float s0 = 0.0f, s1 = 0.0f;
    for (int i = t; i < NT; i += 128) { s0 += colpart[i]; s1 += colpart[NT + i]; }
    sh0[t] = s0; sh1[t] = s1;
    __syncthreads();
    for (int off = 64; off > 0; off >>= 1) {
        if (t < off) { sh0[t] += sh0[t + off]; sh1[t] += sh1[t + off]; }
        __syncthreads();
    }
    if (t == 0) {
        float cx0 = sh0[0] * (1.0f / 6400.0f);
        float cx1 = sh1[0] * (1.0f / 6400.0f);
        out[0] = -0.5f * (__logf(cx0 + 1e-5f) + __logf(cx1 + 1e-5f));
    }
}

extern "C" void kernel_launch(void* const* d_in, const int* in_sizes, int n_in,
                              void* d_out, int out_size, void* d_ws, size_t ws_size,
                              hipStream_t stream) {
    const float* pred   = (const float*)d_in[0];
    const float* target = (const float*)d_in[1];
    float* out = (float*)d_out;

    const size_t optiles = (size_t)NB * NT + OPPAD;   // padded tile records
    float*  W      = (float*)d_ws;
    float*  meanp  = W;                      // 4
    float*  colprt = W + 8;                  // 800
    float*  cq     = W + 816;                // 2*12800 interleaved (c, k/q), 16B aligned
    float2* xop    = (float2*)(W + 816 + 2 * NB * NSP);      // optiles*32 float2
    float2* yop    = xop + optiles * 32;                     // optiles*32 float2

    cl_mean_kernel<<<4, 256, 0, stream>>>(target, meanp);
    cl_normalize_kernel<<<(NB * NSP + 255) / 256, 256, 0, stream>>>(
        pred, target, meanp, xop, yop);
    // One block (4 waves) per 16-wide output tile: B*400 = 800 blocks.
    cl_rowstats_kernel<<<NB * NT, 128, 0, stream>>>(xop, yop, cq);
    cl_rowsum_kernel  <<<NB * NT, 128, 0, stream>>>(xop, yop, cq);
    cl_colmax_kernel  <<<NB * NT, 128, 0, stream>>>(xop, yop, cq, colprt);
    cl_finalize_kernel<<<1, 128, 0, stream>>>(colprt, out);
    (void)in_sizes; (void)n_in; (void)out_size; (void)ws_size;
}